// Location_Sensitive_Stepwise_Monotonic_Attention_1236950581781
// MI455X (gfx1250) — compile-verified
//
#include <hip/hip_runtime.h>
#include <math.h>
#include <float.h>

#define B_ 32
#define T_ 8192
#define RNN_ 1024
#define ENC_ 512
#define ATT_ 128
#define LOC_ 32
#define K_ 31

typedef float v2f __attribute__((ext_vector_type(2)));
typedef float v4f __attribute__((ext_vector_type(4)));
typedef float v8f __attribute__((ext_vector_type(8)));

__device__ __forceinline__ float fast_tanh(float x) {
    // tanh(x) = 1 - 2/(exp(2x)+1); __expf -> v_exp_f32 (TRANS unit)
    float e = __expf(2.0f * x);
    return 1.0f - 2.0f / (e + 1.0f);
}

// ---------------------------------------------------------------------------
// Kernel 0: fold 1x1 into conv:  Wf[a][kk] = sum_l w1[a,l] * wc[l, c, k]
// kk = c*31 + k in [0,62); pad to 64 with zeros (WMMA K padding).
// ---------------------------------------------------------------------------
__global__ void fold_weights(const float* __restrict__ w1,   // [ATT][LOC]
                             const float* __restrict__ wc,   // [LOC][2][K]
                             float* __restrict__ Wf)         // [ATT][64]
{
    int a  = blockIdx.x;     // 0..127
    int kk = threadIdx.x;    // 0..63
    float s = 0.0f;
    if (kk < 2 * K_) {
        int c = kk / K_, k = kk % K_;
        #pragma unroll
        for (int l = 0; l < LOC_; ++l)
            s += w1[a * LOC_ + l] * wc[(l * 2 + c) * K_ + k];
    }
    Wf[a * 64 + kk] = s;
}

// ---------------------------------------------------------------------------
// Kernel 1: q = queries @ w_query.T   (M=B=32, N=ATT=128, K=RNN=1024)
// One wave per 16x16 output tile, fp32 WMMA 16x16x4.
// ---------------------------------------------------------------------------
__global__ __launch_bounds__(32)
void query_gemm(const float* __restrict__ Q,    // [B][RNN]
                const float* __restrict__ Wq,   // [ATT][RNN]
                float* __restrict__ qout)       // [B][ATT]
{
    int l  = threadIdx.x;
    int mt = blockIdx.x >> 3;   // 0..1  (B tiles)
    int nt = blockIdx.x & 7;    // 0..7  (ATT tiles)
    int hi = l >> 4, lo = l & 15;
    int m = mt * 16 + lo;       // batch row
    int n = nt * 16 + lo;       // att col

    v8f acc = {};
    for (int k0 = 0; k0 < RNN_; k0 += 4) {
        int ka = k0 + 2 * hi;
        v2f av, bv;
        av.x = Q[m * RNN_ + ka];
        av.y = Q[m * RNN_ + ka + 1];
        bv.x = Wq[n * RNN_ + ka];       // B[k][n] = w_query[n][k]
        bv.y = Wq[n * RNN_ + ka + 1];
        acc = __builtin_amdgcn_wmma_f32_16x16x4_f32(false, av, false, bv,
                                                    (short)0, acc, false, false);
    }
    #pragma unroll
    for (int i = 0; i < 8; ++i) {
        int mm = mt * 16 + i + 8 * hi;
        qout[mm * ATT_ + n] = acc[i];
    }
}

// ---------------------------------------------------------------------------
// Kernel 2: fused location-conv GEMM + tanh score + sigmoid -> p[b,t]
// 128-thread blocks = 4 waves; each wave owns one (b, 32-column tile).
// loc_att = Wf[128x64] @ window[64x32]: 8 M-tiles x 16 K-steps x 2 N-tiles
// of fp32 WMMA with A-fragments shared across both N-tiles. Fused epilogue:
// +q +pm (non-temporal, read-once), tanh, dot w_score, sigmoid.
// ---------------------------------------------------------------------------
#define TILES_PER_BLOCK 4

__global__ __launch_bounds__(128)
void score_kernel(const float* __restrict__ pm,     // [B][ATT][T]
                  const float* __restrict__ prev,   // [B][T]
                  const float* __restrict__ cum,    // [B][T]
                  const unsigned char* __restrict__ masks, // [B][T] (bool)
                  const float* __restrict__ Wf,     // [ATT][64]
                  const float* __restrict__ qv,     // [B][ATT]
                  const float* __restrict__ wsc,    // [ATT]
                  float* __restrict__ pout)         // [B][T]
{
    __shared__ float sb[TILES_PER_BLOCK][2][64];  // window [t0-15 .. t0+46]

    int w    = threadIdx.x >> 5;                  // wave in block
    int l    = threadIdx.x & 31;
    int hi   = l >> 4, lo = l & 15;

    int tidx = blockIdx.x * TILES_PER_BLOCK + w;  // global 32-col tile id
    int b    = tidx >> 8;                         // T/32 = 256 tiles per batch
    int tile = tidx & 255;
    int t0   = tile * 32;

    const float* ch0 = prev + (size_t)b * T_;
    const float* ch1 = cum  + (size_t)b * T_;
    for (int j = l; j < 62; j += 32) {
        int t = t0 - 15 + j;
        bool ok = (t >= 0) & (t < T_);
        sb[w][0][j] = ok ? ch0[t] : 0.0f;
        sb[w][1][j] = ok ? ch1[t] : 0.0f;
    }
    __syncthreads();

    int tA = t0 + lo;            // N-tile 0 column
    int tB = t0 + 16 + lo;       // N-tile 1 column
    const float* pmb = pm + (size_t)b * ATT_ * T_;
    float part0 = 0.0f, part1 = 0.0f;

    for (int mt = 0; mt < 8; ++mt) {
        v8f acc0 = {}, acc1 = {};
        const float* wrow = Wf + (size_t)(mt * 16 + lo) * 64;  // A row m=lo
        #pragma unroll
        for (int s = 0; s < 16; ++s) {
            int kk = 4 * s + 2 * hi;
            v2f av = *(const v2f*)(wrow + kk);   // Wf[a][kk], Wf[a][kk+1]
            float bx0 = 0.0f, by0 = 0.0f, bx1 = 0.0f, by1 = 0.0f;
            if (kk < 62) {
                int c = kk / K_, k = kk % K_;
                bx0 = sb[w][c][lo + k];
                bx1 = sb[w][c][lo + 16 + k];
            }
            if (kk + 1 < 62) {
                int c = (kk + 1) / K_, k = (kk + 1) % K_;
                by0 = sb[w][c][lo + k];
                by1 = sb[w][c][lo + 16 + k];
            }
            v2f bv0; bv0.x = bx0; bv0.y = by0;
            v2f bv1; bv1.x = bx1; bv1.y = by1;
            acc0 = __builtin_amdgcn_wmma_f32_16x16x4_f32(false, av, false, bv0,
                                                         (short)0, acc0, false, false);
            acc1 = __builtin_amdgcn_wmma_f32_16x16x4_f32(false, av, false, bv1,
                                                         (short)0, acc1, false, false);
        }
        #pragma unroll
        for (int i = 0; i < 8; ++i) {
            int a = mt * 16 + i + 8 * hi;
            float qa = qv[b * ATT_ + a];
            float wa = wsc[a];
            const float* pmr = pmb + (size_t)a * T_;
            float pA = __builtin_nontemporal_load(pmr + tA);  // read-once stream
            float pB = __builtin_nontemporal_load(pmr + tB);
            part0 += wa * fast_tanh(acc0[i] + qa + pA);
            part1 += wa * fast_tanh(acc1[i] + qa + pB);
        }
    }

    // lanes l and l+16 hold the same column: pairwise reduce
    part0 += __shfl_xor(part0, 16, 32);
    part1 += __shfl_xor(part1, 16, 32);
    if (hi == 0) {
        float s0 = part0, s1 = part1;
        if (masks[(size_t)b * T_ + tA]) s0 = -FLT_MAX;
        if (masks[(size_t)b * T_ + tB]) s1 = -FLT_MAX;
        pout[(size_t)b * T_ + tA] = 1.0f / (1.0f + __expf(-s0));
        pout[(size_t)b * T_ + tB] = 1.0f / (1.0f + __expf(-s1));
    }
}

// ---------------------------------------------------------------------------
// Kernel 3: alignments (elementwise, no scan):
//   a[t] = prev[t]*p[t] + (t>0 ? prev[t-1]*(1-p[t-1]) : 0)
// ---------------------------------------------------------------------------
__global__ void align_kernel(const float* __restrict__ prev,
                             const float* __restrict__ pbuf,
                             float* __restrict__ align_out)   // [B][T]
{
    int idx = blockIdx.x * blockDim.x + threadIdx.x;
    if (idx >= B_ * T_) return;
    int t = idx & (T_ - 1);
    float a = prev[idx] * pbuf[idx];
    if (t > 0) a += prev[idx - 1] * (1.0f - pbuf[idx - 1]);
    align_out[idx] = a;
}

// ---------------------------------------------------------------------------
// Kernel 4: contexts[b,c] = sum_t align[b,t] * memories[b,c,t]
// One WAVE per (b,c) row: independent 32 KB float4 streams (non-temporal:
// memories is read exactly once; alignments stay L2-resident and are reused
// 512x per batch). Wave-only reduction, no LDS, no block barrier.
// ---------------------------------------------------------------------------
__global__ __launch_bounds__(256)
void context_kernel(const float* __restrict__ mem,     // [B][ENC][T]
                    const float* __restrict__ alg,     // [B][T]
                    float* __restrict__ ctx)           // [B][ENC]
{
    int w    = threadIdx.x >> 5;
    int lane = threadIdx.x & 31;
    int row  = blockIdx.x * 8 + w;      // 0 .. B*ENC-1
    int b = row >> 9;                   // ENC = 512
    int c = row & (ENC_ - 1);

    const v4f* mrow = (const v4f*)(mem + ((size_t)(b * ENC_ + c)) * T_);
    const v4f* arow = (const v4f*)(alg + (size_t)b * T_);

    float acc = 0.0f;
    #pragma unroll 4
    for (int j = lane; j < T_ / 4; j += 32) {
        if (j + 256 < T_ / 4)
            __builtin_prefetch(&mrow[j + 256], 0, 1);   // global_prefetch_b8
        v4f m4 = __builtin_nontemporal_load(mrow + j);  // read-once stream (TH=NT)
        v4f a4 = arow[j];                               // hot, keep RT policy
        acc += m4.x * a4.x + m4.y * a4.y + m4.z * a4.z + m4.w * a4.w;
    }
    #pragma unroll
    for (int off = 16; off > 0; off >>= 1)
        acc += __shfl_down(acc, off, 32);
    if (lane == 0) ctx[row] = acc;
}

// ---------------------------------------------------------------------------
extern "C" void kernel_launch(void* const* d_in, const int* in_sizes, int n_in,
                              void* d_out, int out_size, void* d_ws, size_t ws_size,
                              hipStream_t stream) {
    const float*         queries    = (const float*)d_in[0];
    const float*         memories   = (const float*)d_in[1];
    const float*         pm         = (const float*)d_in[2];
    const float*         prev       = (const float*)d_in[3];
    const float*         cum        = (const float*)d_in[4];
    const unsigned char* masks      = (const unsigned char*)d_in[5];
    const float*         w_query    = (const float*)d_in[6];
    const float*         w_loc_conv = (const float*)d_in[7];
    const float*         w_loc_1x1  = (const float*)d_in[8];
    const float*         w_score    = (const float*)d_in[9];

    float* ctx_out   = (float*)d_out;            // [B][ENC]
    float* align_out = ctx_out + B_ * ENC_;      // [B][T]

    float* Wf   = (float*)d_ws;                  // 128*64 = 8192 floats
    float* qbuf = Wf + ATT_ * 64;                // 32*128 = 4096 floats
    float* pbuf = qbuf + B_ * ATT_;              // 32*8192 floats

    fold_weights<<<ATT_, 64, 0, stream>>>(w_loc_1x1, w_loc_conv, Wf);
    query_gemm<<<16, 32, 0, stream>>>(queries, w_query, qbuf);

    int n_tiles = B_ * (T_ / 32);                // 8192 wave-tiles
    score_kernel<<<n_tiles / TILES_PER_BLOCK, 32 * TILES_PER_BLOCK, 0, stream>>>(
        pm, prev, cum, masks, Wf, qbuf, w_score, pbuf);

    align_kernel<<<(B_ * T_ + 255) / 256, 256, 0, stream>>>(prev, pbuf, align_out);
    context_kernel<<<(B_ * ENC_) / 8, 256, 0, stream>>>(memories, align_out, ctx_out);
}